// VectorAttention_31516470018462
// MI455X (gfx1250) — compile-verified
//
#include <hip/hip_runtime.h>
#include <math.h>

// Problem constants (from reference): B=2, N=1024, D=32
#define BB 2
#define NN 1024
#define DD 32

typedef __attribute__((ext_vector_type(2))) float v2f;
typedef __attribute__((ext_vector_type(8))) float v8f;

#define WAVES_PER_BLOCK 4          // 128-thread blocks = 4 wave32
#define BLOCKS_X 64
#define WAVES_PER_BATCH (WAVES_PER_BLOCK * BLOCKS_X)   // 256 waves per batch
#define TILES_PER_ROW (NN / 16)                        // 64 tiles of 16 pairs per j-row
#define TILES_PER_BATCH (TILES_PER_ROW * NN)           // 65536 tiles per batch
#define REC_STRIDE 36              // m, Z, acc[32], pad -> 36 floats per wave record
#define LDS_RSTR 34                // row stride (floats): 136B, 8B aligned, bank-spread

// reduce-add across each 16-lane half of a wave32
__device__ __forceinline__ float half16_reduce_add(float t) {
  t += __shfl_xor(t, 1, 32);
  t += __shfl_xor(t, 2, 32);
  t += __shfl_xor(t, 4, 32);
  t += __shfl_xor(t, 8, 32);
  return t;
}

__global__ __launch_bounds__(128, 1) void va_partial(
    const float* __restrict__ P,  const float* __restrict__ V,
    const float* __restrict__ Wv1, const float* __restrict__ bv1,
    const float* __restrict__ Wv2, const float* __restrict__ bv2,
    const float* __restrict__ Ws1, const float* __restrict__ bs1,
    const float* __restrict__ Ws2, const float* __restrict__ bs2,
    float* __restrict__ wsout)
{
  const int b    = blockIdx.y;
  const int lane = threadIdx.x & 31;
  const int wv   = threadIdx.x >> 5;
  const int wid  = blockIdx.x * WAVES_PER_BLOCK + wv;

  const int col0 = lane & 15;        // C-layout column (tile 0)
  const int col1 = col0 + 16;        // C-layout column (tile 1)
  const int hh   = lane >> 4;        // half-wave selector (K / row-half select)
  const int mrow = lane & 15;        // A-layout row owned by this lane

  __shared__ __align__(16) float lds[WAVES_PER_BLOCK][16 * LDS_RSTR];
  float* my = lds[wv];

  // ---- loop-invariant weight fragments (registers) ----
  // WMMA f32 16x16x4: A lane l, elem e -> A[l%16][4k + 2*(l/16) + e]
  //                   B lane l, elem e -> B[4k + 2*(l/16) + e][ntile*16 + l%16]
  v2f bWv2[2][8], bWs1[2][8], wv1a[8], wv1b[8], bv1v[8];
#pragma unroll
  for (int k = 0; k < 8; ++k) {
    const int kk = 4 * k + 2 * hh;
    bWv2[0][k] = v2f{ Wv2[kk*DD + col0], Wv2[(kk+1)*DD + col0] };
    bWv2[1][k] = v2f{ Wv2[kk*DD + col1], Wv2[(kk+1)*DD + col1] };
    bWs1[0][k] = v2f{ Ws1[kk*DD + col0], Ws1[(kk+1)*DD + col0] };
    bWs1[1][k] = v2f{ Ws1[kk*DD + col1], Ws1[(kk+1)*DD + col1] };
    wv1a[k] = v2f{ Wv1[kk],      Wv1[kk+1] };        // Wv1 row 0
    wv1b[k] = v2f{ Wv1[DD + kk], Wv1[DD + kk + 1] }; // Wv1 row 1
    bv1v[k] = v2f{ bv1[kk],      bv1[kk+1] };
  }
  const float bv2_0 = bv2[col0], bv2_1 = bv2[col1];
  const float bs1_0 = bs1[col0], bs1_1 = bs1[col1];
  const float ws2_0 = Ws2[col0], ws2_1 = Ws2[col1];
  const float bs2s  = bs2[0];

  // ---- online softmax state ----
  float m = -INFINITY, Z = 0.f;
  float acc0 = 0.f, acc1 = 0.f;   // per-lane column partials (this half's rows)

  for (int t = wid; t < TILES_PER_BATCH; t += WAVES_PER_BATCH) {
    const int j  = t / TILES_PER_ROW;          // slow pair index (r_j)
    const int i0 = (t % TILES_PER_ROW) * 16;   // fast pair index base (r_i)

    // positions -> invariants for this lane's A-row (both halves redo same row: cheap)
    const float xj = P[(b*NN + j)*3 + 0];
    const float yj = P[(b*NN + j)*3 + 1];
    const float zj = P[(b*NN + j)*3 + 2];
    const int ip = i0 + mrow;
    const float xi = P[(b*NN + ip)*3 + 0];
    const float yi = P[(b*NN + ip)*3 + 1];
    const float zi = P[(b*NN + ip)*3 + 2];

    const float s   = xi*xj + yi*yj + zi*zj;
    const float c01 = xi*yj - yi*xj;
    const float c02 = xi*zj - zi*xj;
    const float c12 = yi*zj - zi*yj;
    const float nb  = sqrtf(c01*c01 + c02*c02 + c12*c12);

    // values in C layout (row r+8*hh at cols col0/col1)
    const float vj0 = V[(b*NN + j)*DD + col0];
    const float vj1 = V[(b*NN + j)*DD + col1];
    float vi0[8], vi1[8];
#pragma unroll
    for (int r = 0; r < 8; ++r) {
      const int row = i0 + r + 8*hh;
      vi0[r] = V[(b*NN + row)*DD + col0];
      vi1[r] = V[(b*NN + row)*DD + col1];
    }

    // ---- matmul 1: h(16x32) @ Wv2(32x32), h built on the fly in A layout ----
    v8f C0 = {}; v8f C1 = {};
#pragma unroll
    for (int k = 0; k < 8; ++k) {
      v2f hk;
      hk.x = fmaxf(0.f, s*wv1a[k].x + nb*wv1b[k].x + bv1v[k].x);
      hk.y = fmaxf(0.f, s*wv1a[k].y + nb*wv1b[k].y + bv1v[k].y);
      C0 = __builtin_amdgcn_wmma_f32_16x16x4_f32(false, hk, false, bWv2[0][k], (short)0, C0, false, false);
      C1 = __builtin_amdgcn_wmma_f32_16x16x4_f32(false, hk, false, bWv2[1][k], (short)0, C1, false, false);
    }

    // joined = 0.5*(invar_values + 0.5*(v_i + v_j)), C layout
    float j0[8], j1[8];
#pragma unroll
    for (int r = 0; r < 8; ++r) {
      j0[r] = 0.5f*(C0[r] + bv2_0) + 0.25f*(vi0[r] + vj0);
      j1[r] = 0.5f*(C1[r] + bv2_1) + 0.25f*(vi1[r] + vj1);
    }

    // ---- C layout -> A layout via wave-private LDS ----
#pragma unroll
    for (int r = 0; r < 8; ++r) {
      my[(r + 8*hh)*LDS_RSTR + col0] = j0[r];
      my[(r + 8*hh)*LDS_RSTR + col1] = j1[r];
    }
    asm volatile("s_wait_dscnt 0" ::: "memory");
    v2f jA[8];
#pragma unroll
    for (int k = 0; k < 8; ++k) {
      const int kk = 4*k + 2*hh;
      const float2 tv = *reinterpret_cast<const float2*>(&my[mrow*LDS_RSTR + kk]);
      jA[k] = v2f{ tv.x, tv.y };
    }

    // ---- matmul 2: joined(16x32) @ Ws1(32x32) ----
    v8f S0 = {}; v8f S1 = {};
#pragma unroll
    for (int k = 0; k < 8; ++k) {
      S0 = __builtin_amdgcn_wmma_f32_16x16x4_f32(false, jA[k], false, bWs1[0][k], (short)0, S0, false, false);
      S1 = __builtin_amdgcn_wmma_f32_16x16x4_f32(false, jA[k], false, bWs1[1][k], (short)0, S1, false, false);
    }

    // ---- scores: per-row dot with Ws2 (half-wave butterfly), then online softmax ----
    float sc[8];
    float tmax = -INFINITY;
#pragma unroll
    for (int r = 0; r < 8; ++r) {
      const float hs0 = fmaxf(0.f, S0[r] + bs1_0);
      const float hs1 = fmaxf(0.f, S1[r] + bs1_1);
      sc[r] = half16_reduce_add(hs0*ws2_0 + hs1*ws2_1) + bs2s;  // score of row r+8*hh
      tmax = fmaxf(tmax, sc[r]);
    }
    tmax = fmaxf(tmax, __shfl_xor(tmax, 16, 32));   // max over all 16 rows (wave-uniform)
    const float mnew  = fmaxf(m, tmax);
    const float scale = __expf(m - mnew);           // exp(-inf)=0 on first tile
    float wsum = 0.f, a0 = 0.f, a1 = 0.f;
#pragma unroll
    for (int r = 0; r < 8; ++r) {
      const float w = __expf(sc[r] - mnew);
      wsum += w;
      a0 += w * j0[r];
      a1 += w * j1[r];
    }
    wsum += __shfl_xor(wsum, 16, 32);               // sum over all 16 rows
    Z    = Z * scale + wsum;
    acc0 = acc0 * scale + a0;
    acc1 = acc1 * scale + a1;
    m = mnew;
  }

  // combine row-halves: full column sums, then write this wave's record
  acc0 += __shfl_xor(acc0, 16, 32);
  acc1 += __shfl_xor(acc1, 16, 32);

  float* rec = wsout + ((size_t)b * WAVES_PER_BATCH + wid) * REC_STRIDE;
  if (lane == 0) { rec[0] = m; rec[1] = Z; }
  if (lane < 16) {
    rec[2 + col0] = acc0;
    rec[2 + col1] = acc1;
  }
}

__global__ __launch_bounds__(32, 1) void va_reduce(const float* __restrict__ wsin,
                                                   float* __restrict__ out)
{
  const int b    = blockIdx.x;
  const int lane = threadIdx.x;
  const float* base = wsin + (size_t)b * WAVES_PER_BATCH * REC_STRIDE;

  // global max
  float mloc = -INFINITY;
  for (int r = lane; r < WAVES_PER_BATCH; r += 32)
    mloc = fmaxf(mloc, base[r*REC_STRIDE + 0]);
#pragma unroll
  for (int d = 1; d < 32; d <<= 1) mloc = fmaxf(mloc, __shfl_xor(mloc, d, 32));
  const float mg = mloc;

  // global Z
  float zloc = 0.f;
  for (int r = lane; r < WAVES_PER_BATCH; r += 32)
    zloc += base[r*REC_STRIDE + 1] * __expf(base[r*REC_STRIDE + 0] - mg);
#pragma unroll
  for (int d = 1; d < 32; d <<= 1) zloc += __shfl_xor(zloc, d, 32);
  const float zg = zloc;

  // lane == output channel (D == warpSize == 32)
  float a = 0.f;
  for (int r = 0; r < WAVES_PER_BATCH; ++r)
    a += __expf(base[r*REC_STRIDE + 0] - mg) * base[r*REC_STRIDE + 2 + lane];
  out[b*DD + lane] = a / zg;
}

extern "C" void kernel_launch(void* const* d_in, const int* in_sizes, int n_in,
                              void* d_out, int out_size, void* d_ws, size_t ws_size,
                              hipStream_t stream) {
  (void)in_sizes; (void)n_in; (void)out_size; (void)ws_size;
  const float* P   = (const float*)d_in[0];
  const float* V   = (const float*)d_in[1];
  const float* Wv1 = (const float*)d_in[2];
  const float* bv1 = (const float*)d_in[3];
  const float* Wv2 = (const float*)d_in[4];
  const float* bv2 = (const float*)d_in[5];
  const float* Ws1 = (const float*)d_in[6];
  const float* bs1 = (const float*)d_in[7];
  const float* Ws2 = (const float*)d_in[8];
  const float* bs2 = (const float*)d_in[9];
  float* ws  = (float*)d_ws;
  float* out = (float*)d_out;

  dim3 grid(BLOCKS_X, BB);
  va_partial<<<grid, 128, 0, stream>>>(P, V, Wv1, bv1, Wv2, bv2, Ws1, bs1, Ws2, bs2, ws);
  va_reduce<<<dim3(BB), 32, 0, stream>>>(ws, out);
}